// FocalLovaszLoss_53017076302494
// MI455X (gfx1250) — compile-verified
//
#include <hip/hip_runtime.h>
#include <cstdint>

// -------- problem constants (match reference) --------
#define EPSV        1e-7f
#define NBINS       4096           // margin in [-1,1] -> 4096 buckets (counting sort)
#define THREADS     256            // 8 wave32s
#define BLOCKS_PER_IMG 32
#define NIMG        32
#define TILE_F      (THREADS * 4)  // floats per tile per tensor (4 KB)

typedef float v2f __attribute__((ext_vector_type(2)));
typedef float v8f __attribute__((ext_vector_type(8)));
typedef int   v4i __attribute__((ext_vector_type(4)));

typedef __attribute__((address_space(1))) v4i GV4;   // global int4
typedef __attribute__((address_space(3))) v4i LV4;   // LDS int4

#if __has_builtin(__builtin_amdgcn_global_load_async_to_lds_b128)
#define HAVE_ASYNC_LDS 1
#else
#define HAVE_ASYNC_LDS 0
#endif

// ---- s_wait_asynccnt <= N ----
__device__ __forceinline__ void wait_async_le2() {
#if __has_builtin(__builtin_amdgcn_s_wait_asynccnt)
    __builtin_amdgcn_s_wait_asynccnt(2);
#else
    asm volatile("s_wait_asynccnt 0x2" ::: "memory");
#endif
}
__device__ __forceinline__ void wait_async_le0() {
#if __has_builtin(__builtin_amdgcn_s_wait_asynccnt)
    __builtin_amdgcn_s_wait_asynccnt(0);
#else
    asm volatile("s_wait_asynccnt 0x0" ::: "memory");
#endif
}

#if HAVE_ASYNC_LDS
// async 16B global -> LDS copy, per lane (tracked by ASYNCcnt)
__device__ __forceinline__ void async_ld16(const float4* g, float* l) {
    __builtin_amdgcn_global_load_async_to_lds_b128(
        (GV4*)(v4i*)(void*)const_cast<float4*>(g),
        (LV4*)(v4i*)(void*)l,
        /*offset=*/0, /*cpol=*/0);
}
#endif

// ---------------------------------------------------------------------------
// Wave-wide f32 sum using the CDNA5 matrix pipe:
//   A(16x4): lane value in VGPR0 -> A[m][0] (lanes 0-15) / A[m][2] (16-31),
//   VGPR1 = 0.  B(4x16) = all ones  =>  D[m][n] = v_m + v_{m+16}  (all n).
//   Per-lane sum of 8 D VGPRs = half-column sum; + shfl_xor(16) = wave sum.
// Exact f32 accumulation (RNE); EXEC all ones at call site (WMMA requirement).
// ---------------------------------------------------------------------------
__device__ __forceinline__ float wave_sum32(float v) {
#if __has_builtin(__builtin_amdgcn_wmma_f32_16x16x4_f32)
    v2f a; a[0] = v;    a[1] = 0.0f;
    v2f b; b[0] = 1.0f; b[1] = 1.0f;
    v8f c = {};
    c = __builtin_amdgcn_wmma_f32_16x16x4_f32(
            false, a, false, b, (short)0, c, false, false);
    float s = c[0] + c[1] + c[2] + c[3] + c[4] + c[5] + c[6] + c[7];
    s += __shfl_xor(s, 16, 32);
    return s;
#else
    for (int off = 16; off > 0; off >>= 1) v += __shfl_xor(v, off, 32);
    return v;
#endif
}

// -------- zero the workspace (histograms + 2 accumulators) --------
__global__ void fl_init(uint32_t* ws, int nwords) {
    int i = blockIdx.x * blockDim.x + threadIdx.x;
    if (i < nwords) ws[i] = 0u;
}

// ---------------------------------------------------------------------------
// Fused streaming pass: focal-loss partial sum + per-image margin histogram.
// grid = (BLOCKS_PER_IMG, NIMG), block = 256. Each block owns a contiguous
// segment of one image. Input tiles are double-buffered into LDS with
// GLOBAL_LOAD_ASYNC_TO_LDS_B128; histogram is LDS-privatized (ds_add_*),
// then merged into a per-image global histogram.
// ---------------------------------------------------------------------------
__global__ void __launch_bounds__(THREADS)
fl_main(const float* __restrict__ x, const float* __restrict__ t,
        uint32_t* __restrict__ gcnt, float* __restrict__ gsab,
        float* __restrict__ acc, int imgElems, int segElems) {
    __shared__ uint32_t s_cnt[NBINS];
    __shared__ float    s_sab[NBINS];
    __shared__ float    s_wv[THREADS / 32];
#if HAVE_ASYNC_LDS
    __shared__ float    s_x[2][TILE_F];
    __shared__ float    s_t[2][TILE_F];
#endif

    const int img = blockIdx.y;
    const int tid = threadIdx.x;

    for (int i = tid; i < NBINS; i += THREADS) { s_cnt[i] = 0u; s_sab[i] = 0.0f; }
    __syncthreads();

    const size_t base = (size_t)img * (size_t)imgElems + (size_t)blockIdx.x * (size_t)segElems;
    const float4* __restrict__ x4 = reinterpret_cast<const float4*>(x + base);
    const float4* __restrict__ t4 = reinterpret_cast<const float4*>(t + base);
    const int iters = segElems / TILE_F;

    float fsum = 0.0f;

#if HAVE_ASYNC_LDS
    // prologue: stage tile 0 into buffer 0
    async_ld16(&x4[tid], &s_x[0][tid * 4]);
    async_ld16(&t4[tid], &s_t[0][tid * 4]);
#endif

    for (int it = 0; it < iters; ++it) {
        float4 xv, tv;
#if HAVE_ASYNC_LDS
        const int cur = it & 1;
        if (it + 1 < iters) {
            const int nxt = cur ^ 1;
            const int j = (it + 1) * THREADS + tid;
            async_ld16(&x4[j], &s_x[nxt][tid * 4]);
            async_ld16(&t4[j], &s_t[nxt][tid * 4]);
            wait_async_le2();      // tile `it` complete; tile `it+1` in flight
        } else {
            wait_async_le0();
        }
        xv = *reinterpret_cast<const float4*>(&s_x[cur][tid * 4]);
        tv = *reinterpret_cast<const float4*>(&s_t[cur][tid * 4]);
#else
        const int i = it * THREADS + tid;
        __builtin_prefetch(x4 + i + 2048, 0, 0);
        __builtin_prefetch(t4 + i + 2048, 0, 0);
        xv = x4[i];
        tv = t4[i];
#endif
        const float xs[4] = {xv.x, xv.y, xv.z, xv.w};
        const float ts[4] = {tv.x, tv.y, tv.z, tv.w};
        #pragma unroll
        for (int k = 0; k < 4; ++k) {
            const float xx = xs[k];
            const float tt = ts[k];
            // sigmoid + clip
            float p = __frcp_rn(1.0f + __expf(-xx));
            p = fminf(fmaxf(p, EPSV), 1.0f - EPSV);
            // stable BCE-with-logits on raw logit
            const float bce = fmaxf(xx, 0.0f) - xx * tt + log1pf(__expf(-fabsf(xx)));
            // focal: alpha=1, gamma=2
            const float pt = (tt > 0.5f) ? p : (1.0f - p);
            const float om = 1.0f - pt;
            fsum = fmaf(om * om, bce, fsum);
            // Lovász margin m = (2p-1)(2t-1) in [-1,1]; bin 0 = largest margin
            const float m = (2.0f * p - 1.0f) * (2.0f * tt - 1.0f);
            int bin = (int)((1.0f - m) * (0.5f * (float)NBINS));
            bin = min(NBINS - 1, max(0, bin));
            atomicAdd(&s_sab[bin], fabsf(m));            // ds_add_f32
            if (tt > 0.5f) atomicAdd(&s_cnt[bin], 1u);   // ds_add_u32
        }
    }
    __syncthreads();

    // merge LDS histogram into per-image global histogram
    uint32_t* gc = gcnt + (size_t)img * NBINS;
    float*    gs = gsab + (size_t)img * NBINS;
    for (int i = tid; i < NBINS; i += THREADS) {
        const uint32_t c = s_cnt[i];
        const float    s = s_sab[i];
        if (c)          atomicAdd(&gc[i], c);
        if (s != 0.0f)  atomicAdd(&gs[i], s);
    }

    // focal partial: WMMA wave reduction -> LDS -> one global atomic per block
    const float wsum = wave_sum32(fsum);        // EXEC all ones here
    const int lane = tid & 31, wid = tid >> 5;
    if (lane == 0) s_wv[wid] = wsum;
    __syncthreads();
    if (tid == 0) {
        float b = 0.0f;
        #pragma unroll
        for (int j = 0; j < THREADS / 32; ++j) b += s_wv[j];
        atomicAdd(&acc[0], b);
    }
}

// ---------------------------------------------------------------------------
// Per-image scan over 4096 bins in descending-margin order (ascending bin):
//   num = sum_k S_k * (P_{<k} + C_k),  denom = total positives.
// One block per image; Hillis-Steele scan over 16 tiles of 256 bins.
// ---------------------------------------------------------------------------
__global__ void __launch_bounds__(THREADS)
fl_scan(const uint32_t* __restrict__ gcnt, const float* __restrict__ gsab,
        float* __restrict__ acc) {
    __shared__ uint32_t sc[THREADS];
    __shared__ double   sd[THREADS];

    const int img = blockIdx.x;
    const int tid = threadIdx.x;
    const uint32_t* gc = gcnt + (size_t)img * NBINS;
    const float*    gs = gsab + (size_t)img * NBINS;

    double   num     = 0.0;
    uint32_t runbase = 0;

    for (int tile = 0; tile < NBINS / THREADS; ++tile) {
        const int i = tile * THREADS + tid;
        const uint32_t c = gc[i];
        const float    S = gs[i];
        sc[tid] = c;
        __syncthreads();
        for (int off = 1; off < THREADS; off <<= 1) {
            const uint32_t v = (tid >= off) ? sc[tid - off] : 0u;
            __syncthreads();
            sc[tid] += v;
            __syncthreads();
        }
        const uint32_t incl = sc[tid];
        num += (double)S * (double)(runbase + incl);
        runbase += sc[THREADS - 1];
        __syncthreads();
    }

    sd[tid] = num;
    __syncthreads();
    for (int off = THREADS / 2; off > 0; off >>= 1) {
        if (tid < off) sd[tid] += sd[tid + off];
        __syncthreads();
    }
    if (tid == 0) {
        const uint32_t denom = runbase;
        const float per = (denom == 0u) ? 0.0f
                          : (float)(sd[0] / (double)fmaxf((float)denom, 1.0f));
        atomicAdd(&acc[1], per);
    }
}

__global__ void fl_final(const float* __restrict__ acc, float* __restrict__ out,
                         float invN) {
    out[0] = fabsf(0.5f * acc[0] * invN) + fabsf(0.5f * acc[1] / (float)NIMG);
}

extern "C" void kernel_launch(void* const* d_in, const int* in_sizes, int n_in,
                              void* d_out, int out_size, void* d_ws, size_t ws_size,
                              hipStream_t stream) {
    const float* x = (const float*)d_in[0];
    const float* t = (const float*)d_in[1];
    float* out = (float*)d_out;

    const int total    = in_sizes[0];            // 32 * 1024 * 1024
    const int imgElems = total / NIMG;           // 1M
    const int segElems = imgElems / BLOCKS_PER_IMG;

    uint32_t* gcnt = (uint32_t*)d_ws;
    float*    gsab = (float*)(gcnt + (size_t)NIMG * NBINS);
    float*    acc  = gsab + (size_t)NIMG * NBINS;   // acc[0]=focal sum, acc[1]=lovasz sum

    const int nwords = NIMG * NBINS * 2 + 2;
    fl_init<<<(nwords + THREADS - 1) / THREADS, THREADS, 0, stream>>>((uint32_t*)d_ws, nwords);

    fl_main<<<dim3(BLOCKS_PER_IMG, NIMG), THREADS, 0, stream>>>(
        x, t, gcnt, gsab, acc, imgElems, segElems);

    fl_scan<<<NIMG, THREADS, 0, stream>>>(gcnt, gsab, acc);

    fl_final<<<1, 1, 0, stream>>>(acc, out, 1.0f / (float)total);
}